// FlashAttnKVCacheModel_32323923870489
// MI455X (gfx1250) — compile-verified
//
#include <hip/hip_runtime.h>
#include <stdint.h>

typedef __attribute__((ext_vector_type(2))) float v2f;
typedef __attribute__((ext_vector_type(4))) float v4f;
typedef __attribute__((ext_vector_type(8))) float v8f;
typedef int v4i __attribute__((vector_size(16)));

#define B_    8
#define H_    32
#define D_    128
#define SMAX_ 4096
#define WAVES 4
#define TILE  16
#define SCALE 0.5f

#if __has_builtin(__builtin_amdgcn_global_load_async_to_lds_b128)
#define HAS_ASYNC 1
#else
#define HAS_ASYNC 0
#endif

typedef __attribute__((address_space(1))) v4i* gvec_t;   // global-AS int4*
typedef __attribute__((address_space(3))) v4i* lvec_t;   // LDS-AS int4*

__device__ __forceinline__ void async_copy16(const float* g, float* l) {
#if HAS_ASYNC
  __builtin_amdgcn_global_load_async_to_lds_b128((gvec_t)g, (lvec_t)l, 0, 0);
#else
  *(v4f*)l = *(const v4f*)g;            // fallback: direct load + ds_store
#endif
}

__device__ __forceinline__ void wait_tile(int have_next) {
#if HAS_ASYNC
  if (have_next) asm volatile("s_wait_asynccnt 0x10" ::: "memory");  // current tile done, next in flight
  else           asm volatile("s_wait_asynccnt 0x0"  ::: "memory");
#else
  asm volatile("" ::: "memory");        // keep ds_store -> ds_load program order
#endif
}

__global__ __launch_bounds__(WAVES * 32)
void fa_decode_kernel(const float* __restrict__ q, const float* __restrict__ k,
                      const float* __restrict__ v, const float* __restrict__ kc,
                      const float* __restrict__ vc, const int* __restrict__ seqlen,
                      float* __restrict__ out) {
  __shared__ float smem[WAVES * 2 * TILE * D_];   // 64 KB: per-wave double-buffered K tiles

  const int bh   = blockIdx.x;
  const int b    = bh >> 5;
  const int h    = bh & (H_ - 1);
  const int tid  = threadIdx.x;
  // wave id is wave-uniform: tell the compiler so all tile/row address math
  // (t, key, kk, row base pointers) scalarizes into SALU + s_cselect,
  // instead of per-lane 64-bit VALU adds + v_cndmask per K row.
  const int wid  = __builtin_amdgcn_readfirstlane(tid >> 5);
  const int lane = tid & 31;

  const int sl = seqlen[b];
  const int L  = sl + 1;                 // keys 0..sl-1 from cache, key sl = new k/v
  const int T  = (L + TILE - 1) / TILE;

  const float*  qrow = q + ((size_t)b * H_ + h) * D_;
  const float*  knew = k + ((size_t)b * H_ + h) * D_;
  const float*  vnew = v + ((size_t)b * H_ + h) * D_;
  const float*  kcb  = kc + (size_t)b * SMAX_ * H_ * D_ + (size_t)h * D_;
  const float*  vcb  = vc + (size_t)b * SMAX_ * H_ * D_ + (size_t)h * D_;
  const size_t  rstride = (size_t)H_ * D_;

  // ---- stage q, build A fragments (16x4 f32 layout: lane m=lane&15, halves hold K pairs) ----
  if (tid < D_) smem[tid] = qrow[tid];
  __syncthreads();

  const int n   = lane & 15;             // B-matrix column / A-matrix row
  const int hlf = lane >> 4;             // K-pair half
  const int dh  = hlf * 2;

  v2f qa[32];
#pragma unroll
  for (int c = 0; c < 32; ++c) {
    float x0 = smem[4 * c + dh];
    float x1 = smem[4 * c + dh + 1];
    bool row0 = (n == 0);                // only M-row 0 carries the query
    qa[c].x = row0 ? x0 : 0.0f;
    qa[c].y = row0 ? x1 : 0.0f;
  }
  __syncthreads();

  float* buf0 = smem + (size_t)(wid * 2    ) * TILE * D_;
  float* buf1 = smem + (size_t)(wid * 2 + 1) * TILE * D_;

  float m_run = -__builtin_inff();
  float l_run = 0.0f;
  v4f   acc   = {0.0f, 0.0f, 0.0f, 0.0f};

  const int t0 = wid;                    // wave handles tiles t0, t0+WAVES, ...
  if (t0 < T) {
    {                                    // prologue: prefetch first tile into buf0
#pragma unroll
      for (int r = 0; r < TILE; ++r) {
        int key = t0 * TILE + r;
        int kk  = key > sl ? sl : key;
        const float* rowp = (kk == sl) ? knew : (kcb + (size_t)kk * rstride);
        async_copy16(rowp + lane * 4, buf0 + r * D_ + lane * 4);
      }
    }
    int it = 0;
    for (int t = t0; t < T; t += WAVES, ++it) {
      const int tn = t + WAVES;
      const int have_next = (tn < T);
      if (have_next) {                   // prefetch next tile into other buffer
        float* dst = (it & 1) ? buf0 : buf1;
#pragma unroll
        for (int r = 0; r < TILE; ++r) {
          int key = tn * TILE + r;
          int kk  = key > sl ? sl : key;
          const float* rowp = (kk == sl) ? knew : (kcb + (size_t)kk * rstride);
          async_copy16(rowp + lane * 4, dst + r * D_ + lane * 4);
        }
      }
      wait_tile(have_next);

      // ---- Q·K^T over this 16-key tile: 32x V_WMMA_F32_16X16X4_F32 ----
      const float* src = (it & 1) ? buf1 : buf0;
      v8f cfrag = {0, 0, 0, 0, 0, 0, 0, 0};
#pragma unroll
      for (int c = 0; c < 32; ++c) {
        v2f bfrag = *(const v2f*)(src + n * D_ + 4 * c + dh);   // K^T fragment from LDS
        cfrag = __builtin_amdgcn_wmma_f32_16x16x4_f32(
            false, qa[c], false, bfrag, (short)0, cfrag, false, false);
      }

      // score for key (t*16+n) lives in cfrag[0] of lanes 0..15 (M=0 row)
      const int keyn = t * TILE + n;
      float s = (hlf == 0 && keyn <= sl) ? cfrag[0] * SCALE : -__builtin_inff();

      // ---- online softmax (wave32 shuffle reductions) ----
      float tmax = s;
#pragma unroll
      for (int off = 16; off > 0; off >>= 1)
        tmax = fmaxf(tmax, __shfl_xor(tmax, off, 32));
      const float m_new = fmaxf(m_run, tmax);
      const float corr  = __expf(m_run - m_new);
      const float p     = __expf(s - m_new);                    // 0 for masked lanes
      float psum = p;
#pragma unroll
      for (int off = 16; off > 0; off >>= 1)
        psum += __shfl_xor(psum, off, 32);
      l_run = l_run * corr + psum;
      m_run = m_new;
      acc *= corr;

      // ---- P·V: broadcast p_j, coalesced 512B V-row loads (scalar base + lane offset) ----
#pragma unroll
      for (int j = 0; j < TILE; ++j) {
        float pj  = __shfl(p, j, 32);
        int   kj  = t * TILE + j;
        int   kkj = kj > sl ? sl : kj;
        const float* vrow = (kkj == sl) ? vnew : (vcb + (size_t)kkj * rstride);
        v4f vv = *(const v4f*)(vrow + lane * 4);
        acc += vv * pj;
      }
    }
  }

  // ---- cross-wave LSE merge (reuse smem; all K-tile reads are complete) ----
  __syncthreads();
  float* red_m = smem;                   // WAVES floats
  float* red_l = smem + 8;               // WAVES floats
  float* red_o = smem + 64;              // WAVES * D_ floats
  if (lane == 0) { red_m[wid] = m_run; red_l[wid] = l_run; }
  *(v4f*)(red_o + wid * D_ + lane * 4) = acc;
  __syncthreads();

  if (tid < D_) {
    float M = -__builtin_inff();
#pragma unroll
    for (int w = 0; w < WAVES; ++w) M = fmaxf(M, red_m[w]);
    float lsum = 0.0f, o = 0.0f;
#pragma unroll
    for (int w = 0; w < WAVES; ++w) {
      float sc = __expf(red_m[w] - M);   // exp(-inf)=0 handles idle waves
      lsum += red_l[w] * sc;
      o    += red_o[w * D_ + tid] * sc;
    }
    out[((size_t)b * H_ + h) * D_ + tid] = o / lsum;
  }
}

extern "C" void kernel_launch(void* const* d_in, const int* in_sizes, int n_in,
                              void* d_out, int out_size, void* d_ws, size_t ws_size,
                              hipStream_t stream) {
  (void)in_sizes; (void)n_in; (void)out_size; (void)d_ws; (void)ws_size;
  const float* q      = (const float*)d_in[0];
  const float* k      = (const float*)d_in[1];
  const float* v      = (const float*)d_in[2];
  const float* kcache = (const float*)d_in[3];
  const float* vcache = (const float*)d_in[4];
  const int*   seqlen = (const int*)d_in[5];
  float*       out    = (float*)d_out;

  dim3 grid(B_ * H_);        // one workgroup per (batch, head)
  dim3 block(WAVES * 32);    // 4 wave32 waves
  hipLaunchKernelGGL(fa_decode_kernel, grid, block, 0, stream,
                     q, k, v, kcache, vcache, seqlen, out);
}